// BayesianSTDPModule_266287973028
// MI455X (gfx1250) — compile-verified
//
#include <hip/hip_runtime.h>

// ---- problem constants ----
#define T_STEPS 2048
#define BATCH   16
#define NIN     1024
#define NOUT    256
#define OSUB    16                 // output rows per workgroup
#define NBLK    (NOUT / OSUB)      // 16 persistent workgroups
#define RSTRIDE (NIN + 4)          // padded LDS row stride: 1028 (1028%64==4 -> conflict-free stride-K fetches)

// LDS layout (in floats)
#define OFF_W     0
#define OFF_PSP   (OFF_W    + OSUB  * RSTRIDE)   // 16448
#define OFF_TPRE  (OFF_PSP  + BATCH * RSTRIDE)   // 32896
#define OFF_ZPRE  (OFF_TPRE + BATCH * RSTRIDE)   // 49344
#define OFF_ZSCR  (OFF_ZPRE + BATCH * RSTRIDE)   // 65792  (8 waves * 32 lanes * 8)
#define OFF_V     (OFF_ZSCR + 8*32*8)            // 67840
#define OFF_ZOUT  (OFF_V    + 256)               // 68096
#define OFF_TPOST (OFF_ZOUT + 256)               // 68352
#define SMEM_FLOATS (OFF_TPOST + 256)            // 68608 floats
#define SMEM_BYTES  (SMEM_FLOATS * sizeof(float)) // 274432 B  (<= 320KB WGP LDS)

// output offsets (floats) in d_out: z_last | v | w | t_pre | t_post
#define OUT_Z     0
#define OUT_V     4096
#define OUT_W     8192
#define OUT_TPRE  270336
#define OUT_TPOST 286720

// ---- async global->LDS path (gfx1250), guarded so the file compiles on any toolchain ----
#if defined(__has_builtin)
#if __has_builtin(__builtin_amdgcn_global_load_async_to_lds_b128) && \
    __has_builtin(__builtin_amdgcn_s_wait_asynccnt)
#define USE_ASYNC_LDS 1
#endif
#endif
#ifndef USE_ASYNC_LDS
#define USE_ASYNC_LDS 0
#endif

typedef float v2f __attribute__((ext_vector_type(2)));
typedef float v8f __attribute__((ext_vector_type(8)));

#if USE_ASYNC_LDS
typedef int i4v __attribute__((__vector_size__(4 * sizeof(int))));
typedef __attribute__((address_space(1))) i4v* g_i4p;   // global int4*
typedef __attribute__((address_space(3))) i4v* l_i4p;   // LDS int4*
#define ASYNC_GPTR(p) ((g_i4p)(unsigned long long)(p))
#define ASYNC_LPTR(p) ((l_i4p)(unsigned)(unsigned long long)(p))
#endif

__global__ __launch_bounds__(256)
void stdp_lif_persistent(const float* __restrict__ spikes,  // [T,B,I]
                         const float* __restrict__ weight,  // [O,I]
                         const float* __restrict__ bias,    // [O]
                         float* __restrict__ out)
{
    extern __shared__ float smem[];
    float* wmat  = smem + OFF_W;     // [OSUB][RSTRIDE]
    float* psp   = smem + OFF_PSP;   // [B][RSTRIDE]
    float* tpre  = smem + OFF_TPRE;  // [B][RSTRIDE]
    float* zpre  = smem + OFF_ZPRE;  // [B][RSTRIDE]
    float* zscr  = smem + OFF_ZSCR;  // [8][32][8] forward partials
    float* vmem  = smem + OFF_V;     // [B][OSUB]
    float* zout  = smem + OFF_ZOUT;  // [B][OSUB]
    float* tpost = smem + OFF_TPOST; // [B][OSUB]

    const int tid   = threadIdx.x;
    const int lane  = tid & 31;
    const int wv    = tid >> 5;      // wave id 0..7
    const int hi    = lane >> 4;     // lane group (K split for A/B operands)
    const int lo    = lane & 15;     // row (A) / col (B,C,D) index
    const int obase = blockIdx.x * OSUB;
    const int kbeg  = wv * (NIN/8);  // this wave's private K/column range [kbeg, kbeg+128)
    const int col   = kbeg + lane*4; // this lane's float4 column within the wave range

    // ---------------- init state in LDS ----------------
    for (int r = 0; r < OSUB; ++r)
        *(float4*)&wmat[r*RSTRIDE + tid*4] =
            *(const float4*)&weight[(size_t)(obase + r)*NIN + tid*4];
    {
        float4 z4 = {0.f,0.f,0.f,0.f};
        float4 o4 = {1.f,1.f,1.f,1.f};
        for (int r = 0; r < BATCH; ++r) {
            *(float4*)&psp [r*RSTRIDE + tid*4] = z4;
            *(float4*)&tpre[r*RSTRIDE + tid*4] = o4;   // t_pre0 = 1
            *(float4*)&zpre[r*RSTRIDE + tid*4] = z4;
        }
    }
    vmem[tid]  = 0.f;
    zout[tid]  = 0.f;
    tpost[tid] = 1.f;                                   // t_post0 = 1
    const float bias_r = bias[obase + (tid & 15)];
    __syncthreads();

    // ---------------- time loop (3 barriers / step) ----------------
    for (int t = 0; t < T_STEPS; ++t) {
        // -------- phase 1: psp / t_pre / z_pre update, wave-private columns --------
        // Each wave updates exactly the columns its own forward-GEMM chunk reads, so
        // no barrier is needed between phase 1 and phase 2 (same-wave DS ops are in order).
        const float* sp = spikes + (size_t)t * (BATCH*NIN);
        #pragma unroll 4
        for (int r = 0; r < BATCH; ++r) {
            const int gi = r*NIN     + col;
            const int l  = r*RSTRIDE + col;
#if USE_ASYNC_LDS
            // DMA the raw spike slice into z_pre; overlaps with phases 1-3,
            // consumed in phase 4 after s_wait_asynccnt + barrier.
            __builtin_amdgcn_global_load_async_to_lds_b128(
                ASYNC_GPTR(&sp[gi]), ASYNC_LPTR(&zpre[l]), 0, 0);
#endif
            float4 s4 = *(const float4*)&sp[gi];
            float4 p4 = *(float4*)&psp[l];
            p4.x = 0.95f*p4.x + s4.x;  p4.y = 0.95f*p4.y + s4.y;
            p4.z = 0.95f*p4.z + s4.z;  p4.w = 0.95f*p4.w + s4.w;
            *(float4*)&psp[l] = p4;
            float4 q4 = *(float4*)&tpre[l];
            q4.x = 0.95f*q4.x + 0.05f*s4.x;  q4.y = 0.95f*q4.y + 0.05f*s4.y;
            q4.z = 0.95f*q4.z + 0.05f*s4.z;  q4.w = 0.95f*q4.w + 0.05f*s4.w;
            *(float4*)&tpre[l] = q4;
#if !USE_ASYNC_LDS
            *(float4*)&zpre[l] = s4;
#endif
            if (t + 1 < T_STEPS)
                __builtin_prefetch(&sp[BATCH*NIN + gi], 0, 1);
        }

        // -------- phase 2: forward GEMM  z_in = psp @ w^T  (K split over waves) --------
        // A = psp tile (M=b, K), B = w^T tile (K, N=o).  f32 WMMA, K=4 per instruction.
        v8f acc = {0.f,0.f,0.f,0.f,0.f,0.f,0.f,0.f};
        #pragma unroll
        for (int k4 = 0; k4 < NIN/8; k4 += 4) {
            const int kk = kbeg + k4 + 2*hi;
            v2f a = *(const v2f*)&psp [lo*RSTRIDE + kk];   // A[m=lo][kk,kk+1]
            v2f b = *(const v2f*)&wmat[lo*RSTRIDE + kk];   // B[kk,kk+1][n=lo] = w[lo][kk..]
            acc = __builtin_amdgcn_wmma_f32_16x16x4_f32(false, a, false, b,
                                                        (short)0, acc, false, false);
        }
        *(v8f*)&zscr[(wv*32 + lane)*8] = acc;              // per-lane contiguous partial
        __syncthreads();

        // -------- phase 3: reduce partials + LIF + t_post --------
        {
            const int b = tid >> 4, o = tid & 15;          // D element (m=b, n=o)
            const int ln = ((b >> 3) << 4) + o;            // lane holding it
            const int rr = b & 7;                          // vgpr index
            float sum = bias_r;
            #pragma unroll
            for (int w8 = 0; w8 < 8; ++w8)
                sum += zscr[(w8*32 + ln)*8 + rr];
            float v = vmem[tid];
            v = v + 0.1f*(sum - v);                        // dt*tau_mem_inv = 0.1
            float z = (v >= 1.0f) ? 1.0f : 0.0f;
            v = (z > 0.f) ? 0.f : v;
            vmem[tid] = v;
            zout[tid] = z;
            float q = tpost[tid];
            tpost[tid] = q + 0.05f*(z - q);                // dt*tau_post_inv = 0.05
        }
#if USE_ASYNC_LDS
        __builtin_amdgcn_s_wait_asynccnt(0);               // z_pre DMA landed in LDS
#endif
        __syncthreads();

        // -------- phase 4: fused STDP GEMM  w = clip(w + A*B), M=OSUB, K=2B=32 --------
        // A[o][k] = k<16 ?  eta+ * z_out[k][o]  :  -eta- * t_post[k-16][o]
        // B[k][i] = k<16 ?  t_pre[k][i]         :   z_pre[k-16][i]
        v2f aw[8];
        #pragma unroll
        for (int kq = 0; kq < 8; ++kq) {
            const int k = kq*4 + 2*hi;                     // pair (k,k+1), uniform region per kq
            v2f a;
            if (k < BATCH) {
                a.x =  1.0e-3f * zout[(k  )*OSUB + lo];
                a.y =  1.0e-3f * zout[(k+1)*OSUB + lo];
            } else {
                a.x = -1.0e-3f * tpost[(k-16)*OSUB + lo];
                a.y = -1.0e-3f * tpost[(k-15)*OSUB + lo];
            }
            aw[kq] = a;
        }
        #pragma unroll
        for (int tile = 0; tile < 8; ++tile) {
            const int i0 = (tile*8 + wv) * 16;             // this wave's i-tiles
            v8f c;                                          // C = current weight tile
            #pragma unroll
            for (int r = 0; r < 8; ++r)
                c[r] = wmat[(r + 8*hi)*RSTRIDE + i0 + lo];
            #pragma unroll
            for (int kq = 0; kq < 8; ++kq) {
                const int k = kq*4 + 2*hi;
                v2f b;
                if (k < BATCH) {
                    b.x = tpre[(k  )*RSTRIDE + i0 + lo];
                    b.y = tpre[(k+1)*RSTRIDE + i0 + lo];
                } else {
                    b.x = zpre[(k-16)*RSTRIDE + i0 + lo];
                    b.y = zpre[(k-15)*RSTRIDE + i0 + lo];
                }
                c = __builtin_amdgcn_wmma_f32_16x16x4_f32(false, aw[kq], false, b,
                                                          (short)0, c, false, false);
            }
            #pragma unroll
            for (int r = 0; r < 8; ++r) {                  // clamp to [w_min, w_max]
                float x = c[r];
                x = fminf(fmaxf(x, 0.0f), 1.0f);
                wmat[(r + 8*hi)*RSTRIDE + i0 + lo] = x;
            }
        }
        __syncthreads();                                    // w visible to next step's phase 2
    }

    // ---------------- write outputs ----------------
    {
        const int b = tid >> 4, o = tid & 15;
        const int og = obase + o;
        out[OUT_Z     + b*NOUT + og] = zout[tid];
        out[OUT_V     + b*NOUT + og] = vmem[tid];
        out[OUT_TPOST + b*NOUT + og] = tpost[tid];
        for (int r = 0; r < OSUB; ++r)
            *(float4*)&out[OUT_W + (size_t)(obase + r)*NIN + tid*4] =
                *(float4*)&wmat[r*RSTRIDE + tid*4];
        if (blockIdx.x == 0) {
            for (int r = 0; r < BATCH; ++r)
                *(float4*)&out[OUT_TPRE + r*NIN + tid*4] =
                    *(float4*)&tpre[r*RSTRIDE + tid*4];
        }
    }
}

extern "C" void kernel_launch(void* const* d_in, const int* in_sizes, int n_in,
                              void* d_out, int out_size, void* d_ws, size_t ws_size,
                              hipStream_t stream)
{
    (void)in_sizes; (void)n_in; (void)out_size; (void)d_ws; (void)ws_size;
    const float* spikes = (const float*)d_in[0];   // [T,B,I] f32
    const float* weight = (const float*)d_in[1];   // [O,I]  f32
    const float* bias   = (const float*)d_in[2];   // [O]    f32
    float* out = (float*)d_out;

    (void)hipFuncSetAttribute(reinterpret_cast<const void*>(&stdp_lif_persistent),
                              hipFuncAttributeMaxDynamicSharedMemorySize,
                              (int)SMEM_BYTES);
    stdp_lif_persistent<<<NBLK, 256, SMEM_BYTES, stream>>>(spikes, weight, bias, out);
}